// TriplesDistances_16234976379049
// MI455X (gfx1250) — compile-verified
//
#include <hip/hip_runtime.h>
#include <hip/hip_bf16.h>
#include <stdint.h>

// Problem shape (fixed by the reference's setup_inputs()).
#define B_      16
#define N_      512
#define A_      1024
#define NPB     4        // rows of n handled per block
#define THREADS 256      // 8 wave32s per block

// positions[b] table: 512 * 3 floats = 6144 bytes = 384 x 16B transfers
#define STAGE_B128 ((N_ * 3 * 4) / 16)   // 384

typedef long long ll2 __attribute__((ext_vector_type(2)));
typedef float     f2  __attribute__((ext_vector_type(2)));

__device__ __forceinline__ void async_b128_to_lds(unsigned laddr, unsigned long long gaddr) {
    asm volatile("global_load_async_to_lds_b128 %0, %1, off"
                 :: "v"(laddr), "v"(gaddr)
                 : "memory");
}

__device__ __forceinline__ float safe_dist(float dx, float dy, float dz) {
    float s = dx * dx + dy * dy + dz * dz;
    return (s > 0.0f) ? __builtin_sqrtf(s) : 0.0f;
}

__global__ __launch_bounds__(THREADS) void TriplesDistances_kernel(
    const float* __restrict__ pos,        // [B, N, 3] f32
    const long long* __restrict__ nj,     // [B, N, A] i64
    const long long* __restrict__ nk,     // [B, N, A] i64
    float* __restrict__ out)              // [3, B, N, A] f32 (r_ij | r_ik | r_jk)
{
    __shared__ float sp[N_ * 3];          // 6 KB per-batch position table

    const int blocksPerB = N_ / NPB;                  // 128
    const int b  = blockIdx.x / blocksPerB;
    const int n0 = (blockIdx.x % blocksPerB) * NPB;

    // ---- Stage positions[b] (6 KB) into LDS via gfx1250 async global->LDS DMA.
    // 384 x b128 transfers with 256 threads: fully unrolled (1 full + 1 half pass).
    {
        const unsigned long long gsrc =
            (unsigned long long)(uintptr_t)(pos + (size_t)b * (N_ * 3));
        // addrspace(3) flat address: low 32 bits are the LDS byte offset
        const unsigned lbase = (unsigned)(uintptr_t)(&sp[0]);
        const unsigned t16   = (unsigned)threadIdx.x * 16u;

        async_b128_to_lds(lbase + t16, gsrc + t16);
        if (threadIdx.x < (STAGE_B128 - THREADS)) {   // threads 0..127
            async_b128_to_lds(lbase + t16 + THREADS * 16u,
                              gsrc + t16 + THREADS * 16u);
        }
        asm volatile("s_wait_asynccnt 0x0" ::: "memory");
    }
    __syncthreads();

    const size_t BNA = (size_t)B_ * N_ * A_;
    float* __restrict__ out_ij = out;
    float* __restrict__ out_ik = out + BNA;
    float* __restrict__ out_jk = out + 2 * BNA;

    for (int nl = 0; nl < NPB; ++nl) {
        const int n = n0 + nl;
        const float ix = sp[3 * n + 0];
        const float iy = sp[3 * n + 1];
        const float iz = sp[3 * n + 2];

        const size_t rowbase = ((size_t)b * N_ + n) * A_;
        const long long* __restrict__ njr = nj + rowbase;
        const long long* __restrict__ nkr = nk + rowbase;

        // 2 consecutive elements per thread per pass: 16B index loads, 8B stores.
#pragma unroll
        for (int a0 = (int)threadIdx.x * 2; a0 < A_; a0 += THREADS * 2) {
            const ll2 vj = __builtin_nontemporal_load((const ll2*)&njr[a0]);
            const ll2 vk = __builtin_nontemporal_load((const ll2*)&nkr[a0]);
            const int j0 = (int)vj.x, j1 = (int)vj.y;
            const int k0 = (int)vk.x, k1 = (int)vk.y;

            // LDS gathers (random within 6 KB table)
            const float j0x = sp[3 * j0 + 0], j0y = sp[3 * j0 + 1], j0z = sp[3 * j0 + 2];
            const float j1x = sp[3 * j1 + 0], j1y = sp[3 * j1 + 1], j1z = sp[3 * j1 + 2];
            const float k0x = sp[3 * k0 + 0], k0y = sp[3 * k0 + 1], k0z = sp[3 * k0 + 2];
            const float k1x = sp[3 * k1 + 0], k1y = sp[3 * k1 + 1], k1z = sp[3 * k1 + 2];

            f2 rij, rik, rjk;
            rij.x = safe_dist(j0x - ix, j0y - iy, j0z - iz);
            rij.y = safe_dist(j1x - ix, j1y - iy, j1z - iz);
            rik.x = safe_dist(k0x - ix, k0y - iy, k0z - iz);
            rik.y = safe_dist(k1x - ix, k1y - iy, k1z - iz);
            rjk.x = safe_dist(j0x - k0x, j0y - k0y, j0z - k0z);
            rjk.y = safe_dist(j1x - k1x, j1y - k1y, j1z - k1z);

            // streaming, single-use writes -> non-temporal b64 stores
            const size_t o0 = rowbase + (size_t)a0;
            __builtin_nontemporal_store(rij, (f2*)&out_ij[o0]);
            __builtin_nontemporal_store(rik, (f2*)&out_ik[o0]);
            __builtin_nontemporal_store(rjk, (f2*)&out_jk[o0]);
        }
    }
}

extern "C" void kernel_launch(void* const* d_in, const int* in_sizes, int n_in,
                              void* d_out, int out_size, void* d_ws, size_t ws_size,
                              hipStream_t stream) {
    (void)in_sizes; (void)n_in; (void)out_size; (void)d_ws; (void)ws_size;
    const float*     pos = (const float*)d_in[0];
    const long long* nj  = (const long long*)d_in[1];
    const long long* nk  = (const long long*)d_in[2];
    float*           out = (float*)d_out;

    dim3 grid(B_ * (N_ / NPB));   // 2048 blocks
    dim3 block(THREADS);          // 8 wave32s
    TriplesDistances_kernel<<<grid, block, 0, stream>>>(pos, nj, nk, out);
}